// rgb_Aggregate_30116310680031
// MI455X (gfx1250) — compile-verified
//
#include <hip/hip_runtime.h>

typedef float v2f __attribute__((ext_vector_type(2)));
typedef float v8f __attribute__((ext_vector_type(8)));

#define HDIM   300
#define WDIM   300
#define PS     5
#define PPR    60      // patches per row
#define LPAT   3600    // patches per image
#define CHW    90000   // H*W
#define NC     32      // channels per branch image
#define KW     1152    // 128*9 weight row length
#define TILE_W 32      // output pixels per block (along a row)
#define TPR    10      // col tiles per row (10*32 >= 300)
#define BCOLS  34      // TILE_W + 2 halo cols
#define CPAD   132     // padded ci stride (128 -> 132, breaks 64-bank aliasing)

__global__ __launch_bounds__(128) void fused_gather_conv(
    const float* __restrict__ x_rgb, const float* __restrict__ y_hsi,
    const int*   __restrict__ corr,
    const float* __restrict__ rgb_gamma, const float* __restrict__ hsi_gamma,
    const float* __restrict__ rgb_W, const float* __restrict__ rgb_b,
    const float* __restrict__ hsi_W, const float* __restrict__ hsi_b,
    float* __restrict__ out)
{
  // B panel: gathered 128-ch conv input, [halo_row][halo_col][ci] (ci padded to 132)
  __shared__ float ldsB[3 * BCOLS * CPAD];
  // A panel: weights for current 3x3 tap t, [co][ci]
  __shared__ float ldsA[NC * CPAD];

  const int tid = threadIdx.x;
  int b = blockIdx.x;
  const int tw     = b % TPR;   b /= TPR;
  const int h      = b % HDIM;  b /= HDIM;
  const int n      = b % 4;     b /= 4;
  const int branch = b;         // 0 = rgb branch, 1 = hsi branch
  const int w0 = tw * TILE_W;

  // Source-channel mapping of the concatenated branch image:
  //  rgb: ch 0..15 -> x_rgb[16+cc], ch 16..31 -> y_hsi[cc]
  //  hsi: ch 0..15 -> x_rgb[cc],    ch 16..31 -> y_hsi[cc-16]
  const float* srcLo = x_rgb + (size_t)n * NC * CHW + (branch == 0 ? 16 * CHW : 0);
  const float* srcHi = y_hsi + (size_t)n * NC * CHW - (branch == 1 ? 16 * CHW : 0);
  const float* gamp  = branch ? hsi_gamma : rgb_gamma;
  const float* Wg    = branch ? hsi_W : rgb_W;
  const float* bias  = branch ? hsi_b : rgb_b;

  // ---------- Phase 1: gather B panel (patch-shuffled, gamma-scaled) into LDS ----------
  if (tid < 3 * BCOLS) {
    const int r  = tid / BCOLS;            // halo row 0..2 (kh)
    const int c  = tid % BCOLS;            // halo col 0..33
    const int hh = h  + r - 1;
    const int ww = w0 + c - 1;
    float* dst = &ldsB[(r * BCOLS + c) * CPAD];
    const bool valid = (hh >= 0) & (hh < HDIM) & (ww >= 0) & (ww < WDIM);
    int sj0 = 0, sj1 = 0, sj2 = 0, sj3 = 0, ph = 0, pw = 0;
    if (valid) {
      const int bh = hh / PS, bw = ww / PS;
      ph = hh - bh * PS; pw = ww - bw * PS;
      const int j = bh * PPR + bw;
      const int4 id = *(const int4*)(corr + ((size_t)n * LPAT + j) * 4); // corr[0][n][j][0..3]
      sj0 = j; sj1 = id.y; sj2 = id.z; sj3 = id.w;   // slot 0 is patch j itself
    }
    const int sj[4] = {sj0, sj1, sj2, sj3};
    #pragma unroll
    for (int kb = 0; kb < 4; ++kb) {
      const int s   = sj[kb];
      const int sr  = s / PPR;
      const int off = (sr * PS + ph) * WDIM + (s - sr * PPR) * PS + pw;
      const float g = gamp[kb];
      #pragma unroll
      for (int cc = 0; cc < NC; cc += 4) {
        const float* sp = (cc < 16) ? srcLo : srcHi;
        float4 v;
        v.x = valid ? sp[(cc + 0) * CHW + off] : 0.f;
        v.y = valid ? sp[(cc + 1) * CHW + off] : 0.f;
        v.z = valid ? sp[(cc + 2) * CHW + off] : 0.f;
        v.w = valid ? sp[(cc + 3) * CHW + off] : 0.f;
        v.x *= g; v.y *= g; v.z *= g; v.w *= g;
        *(float4*)&dst[kb * NC + cc] = v;
      }
    }
  }
  __syncthreads();

  // ---------- Phase 2: implicit-GEMM conv via V_WMMA_F32_16X16X4_F32 ----------
  const int lane = tid & 31;
  const int wv   = tid >> 5;
  const int half = lane >> 4;    // 0: K=0..1 rows, 1: K=2..3 rows (A/B frag split)
  const int ln   = lane & 15;
  const int co0  = (wv & 1) * 16;     // output-channel tile base
  const int pix0 = (wv >> 1) * 16;    // pixel tile base
  const int p    = pix0 + ln;         // this lane's pixel column within tile

  const int sco = tid >> 2;           // weight staging: co 0..31
  const int sci = (tid & 3) * 32;     // weight staging: ci block

  v8f acc = {0.f, 0.f, 0.f, 0.f, 0.f, 0.f, 0.f, 0.f};

  for (int t = 0; t < 9; ++t) {
    // stage weights for tap t: ldsA[co][ci] = Wg[co][ci*9 + t]
    {
      const float* wrow = Wg + (size_t)sco * KW + t;
      #pragma unroll
      for (int u = 0; u < 32; u += 4) {
        float4 a;
        a.x = wrow[(sci + u + 0) * 9];
        a.y = wrow[(sci + u + 1) * 9];
        a.z = wrow[(sci + u + 2) * 9];
        a.w = wrow[(sci + u + 3) * 9];
        *(float4*)&ldsA[sco * CPAD + sci + u] = a;
      }
    }
    __syncthreads();

    const int kh  = t / 3;
    const int kwp = t - kh * 3;
    // B fragment source: row kh, col p+kwp of halo, channels ci..ci+3
    const float* brow = &ldsB[(kh * BCOLS + (p + kwp)) * CPAD + half * 2];
    // A fragment source: row co0+ln, channels ci..ci+3
    const float* arow = &ldsA[(co0 + ln) * CPAD + half * 2];

    #pragma unroll
    for (int cib = 0; cib < 32; ++cib) {
      const v2f av = *(const v2f*)(arow + cib * 4);  // A[16x4]: lanes<16 K0,K1; lanes>=16 K2,K3
      const v2f bv = *(const v2f*)(brow + cib * 4);  // B[4x16]: same K split, N = ln
#if __has_builtin(__builtin_amdgcn_wmma_f32_16x16x4_f32)
      acc = __builtin_amdgcn_wmma_f32_16x16x4_f32(false, av, false, bv,
                                                  (short)0, acc, false, false);
#else
      // placeholder (should not trigger; histogram wmma==0 would flag it)
      acc[0] += av[0] * bv[0] + av[1] * bv[1];
#endif
    }
    __syncthreads();
  }

  // ---------- Epilogue: bias + residual + predicated store ----------
  const int wcol = w0 + p;
  if (wcol < WDIM) {
    float* outp = out + (size_t)branch * (4u * NC * CHW)   // out_rgb | out_hsi
                      + (size_t)n * NC * CHW + h * WDIM + wcol;
    #pragma unroll
    for (int r = 0; r < 8; ++r) {
      const int co = co0 + half * 8 + r;        // C/D layout: lanes>=16 hold rows M=r+8
      const float* sp = (co < 16) ? srcLo : srcHi;
      const float resv = sp[co * CHW + h * WDIM + wcol];
      outp[(size_t)co * CHW] = acc[r] + bias[co] + resv;
    }
  }
}

extern "C" void kernel_launch(void* const* d_in, const int* in_sizes, int n_in,
                              void* d_out, int out_size, void* d_ws, size_t ws_size,
                              hipStream_t stream) {
  const float* x_rgb     = (const float*)d_in[0];
  const float* y_hsi     = (const float*)d_in[1];
  const int*   corr      = (const int*)d_in[2];
  const float* rgb_gamma = (const float*)d_in[3];
  const float* hsi_gamma = (const float*)d_in[4];
  const float* rgb_W     = (const float*)d_in[5];
  const float* rgb_b     = (const float*)d_in[6];
  const float* hsi_W     = (const float*)d_in[7];
  const float* hsi_b     = (const float*)d_in[8];
  float* out = (float*)d_out;

  const int nblocks = 2 * 4 * HDIM * TPR;   // branches * N * rows * col-tiles = 24000
  fused_gather_conv<<<dim3(nblocks), dim3(128), 0, stream>>>(
      x_rgb, y_hsi, corr, rgb_gamma, hsi_gamma, rgb_W, rgb_b, hsi_W, hsi_b, out);
}